// Decoder_37056977829937
// MI455X (gfx1250) — compile-verified
//
#include <hip/hip_runtime.h>
#include <math.h>

// ---------------------------------------------------------------------------
// Tacotron decoder step for MI455X (gfx1250, wave32, WMMA).
// All GEMMs use V_WMMA_F32_16X16X4_F32 (fp32 in/out -> reference-accurate).
// Each wave computes a 16x32 C tile (2 accumulators share one A fragment).
// Attention scans (2 x 128MB) are bandwidth-bound streaming kernels with
// hardware-transcendental tanh (v_exp_f32 + v_rcp_f32).
// ---------------------------------------------------------------------------

#define Bz    128
#define Tt    1024
#define Dd    256
#define Ll    512
#define NMELS 80
#define MAXR  20

typedef float v2f __attribute__((ext_vector_type(2)));
typedef float v8f __attribute__((ext_vector_type(8)));

__device__ __forceinline__ float sigmoidf_(float x) { return 1.0f / (1.0f + expf(-x)); }

// fast tanh: (e^2x - 1) / (e^2x + 1) with hardware exp/rcp; clamp avoids inf/inf
__device__ __forceinline__ float tanh_fast(float x) {
    float xc = fminf(fmaxf(x, -10.0f), 10.0f);
    float t  = __expf(2.0f * xc);
    return (t - 1.0f) * __builtin_amdgcn_rcpf(t + 1.0f);
}

// Output region offsets (depend on runtime r):
// [mels B*80*r][scores B*T][attn_h B*256][rnn1_h B*512][rnn2_h][rnn1_c][rnn2_c][context B*256]
__device__ __forceinline__ size_t off_scores (int r) { return (size_t)Bz * NMELS * r; }
__device__ __forceinline__ size_t off_attnh  (int r) { return off_scores(r) + (size_t)Bz * Tt; }
__device__ __forceinline__ size_t off_hc     (int r) { return off_attnh(r)  + (size_t)Bz * Dd; }
__device__ __forceinline__ size_t off_ctx    (int r) { return off_hc(r)     + (size_t)4 * Bz * Ll; }

// ---------------------------------------------------------------------------
// Generic fp32 WMMA GEMM:  C = act( A0 @ W0^T + A1 @ W1^T + bias0 + bias1 )
// A row-major (M x K, row stride lda), W row-major (N x K, row stride ldw).
// One wave computes one 16x32 tile of C (two V_WMMA_F32_16X16X4_F32 per k-step,
// sharing the A fragment). N must be a multiple of 32.
// ---------------------------------------------------------------------------
__global__ __launch_bounds__(32)
void wmma_gemm2_kernel(const float* __restrict__ A0, int lda0,
                       const float* __restrict__ W0, int ldw0, int K0,
                       const float* __restrict__ A1, int lda1,
                       const float* __restrict__ W1, int ldw1, int K1,
                       const float* __restrict__ bias0,
                       const float* __restrict__ bias1,
                       float* __restrict__ C, int ldc, int relu)
{
    const int lane = threadIdx.x;   // 0..31 (wave32)
    const int l15  = lane & 15;
    const int half = lane >> 4;

    const int n0 = blockIdx.x * 32;       // two 16-wide column tiles
    const int m0 = blockIdx.y * 16;

    float b0i = 0.0f, b1i = 0.0f;
    if (bias0) { b0i += bias0[n0 + l15]; b1i += bias0[n0 + 16 + l15]; }
    if (bias1) { b0i += bias1[n0 + l15]; b1i += bias1[n0 + 16 + l15]; }

    v8f c0, c1;
    #pragma unroll
    for (int i = 0; i < 8; ++i) { c0[i] = b0i; c1[i] = b1i; }

    // A fragment: lane holds A[m][k + 2*half .. +1]; B fragment: W[n][k + 2*half .. +1]
    {
        const float* ap  = A0 + (size_t)(m0 + l15) * lda0 + 2 * half;
        const float* bp0 = W0 + (size_t)(n0 + l15) * ldw0 + 2 * half;
        const float* bp1 = W0 + (size_t)(n0 + 16 + l15) * ldw0 + 2 * half;
        #pragma unroll 8
        for (int k = 0; k < K0; k += 4) {
            v2f a  = *(const v2f*)(ap + k);
            v2f b0 = *(const v2f*)(bp0 + k);
            v2f b1 = *(const v2f*)(bp1 + k);
            c0 = __builtin_amdgcn_wmma_f32_16x16x4_f32(false, a, false, b0,
                                                       (short)0, c0, false, false);
            c1 = __builtin_amdgcn_wmma_f32_16x16x4_f32(false, a, false, b1,
                                                       (short)0, c1, false, false);
        }
    }
    if (K1 > 0) {
        const float* ap  = A1 + (size_t)(m0 + l15) * lda1 + 2 * half;
        const float* bp0 = W1 + (size_t)(n0 + l15) * ldw1 + 2 * half;
        const float* bp1 = W1 + (size_t)(n0 + 16 + l15) * ldw1 + 2 * half;
        #pragma unroll 8
        for (int k = 0; k < K1; k += 4) {
            v2f a  = *(const v2f*)(ap + k);
            v2f b0 = *(const v2f*)(bp0 + k);
            v2f b1 = *(const v2f*)(bp1 + k);
            c0 = __builtin_amdgcn_wmma_f32_16x16x4_f32(false, a, false, b0,
                                                       (short)0, c0, false, false);
            c1 = __builtin_amdgcn_wmma_f32_16x16x4_f32(false, a, false, b1,
                                                       (short)0, c1, false, false);
        }
    }

    // C layout: VGPR r -> row (r + 8*half), col l15 (+0 / +16)
    #pragma unroll
    for (int rr = 0; rr < 8; ++rr) {
        float v0 = c0[rr], v1 = c1[rr];
        if (relu) { v0 = fmaxf(v0, 0.0f); v1 = fmaxf(v1, 0.0f); }
        float* crow = C + (size_t)(m0 + rr + 8 * half) * ldc + n0;
        crow[l15]      = v0;
        crow[16 + l15] = v1;
    }
}

// ---------------------------------------------------------------------------
// GRU gate fusion: attn_h = (1-z)*n + z*h_prev
// ---------------------------------------------------------------------------
__global__ __launch_bounds__(256)
void gru_gate_kernel(const float* __restrict__ gi, const float* __restrict__ gh,
                     const float* __restrict__ h_prev,
                     float* __restrict__ attn_h_s, float* __restrict__ out,
                     const int* __restrict__ rp)
{
    int idx = blockIdx.x * blockDim.x + threadIdx.x;   // b*256 + d
    if (idx >= Bz * Dd) return;
    int b = idx >> 8, d = idx & 255;
    const float* gib = gi + (size_t)b * (3 * Dd);
    const float* ghb = gh + (size_t)b * (3 * Dd);
    float rg = sigmoidf_(gib[d]          + ghb[d]);
    float zg = sigmoidf_(gib[Dd + d]     + ghb[Dd + d]);
    float ng = tanhf    (gib[2 * Dd + d] + rg * ghb[2 * Dd + d]);
    float h  = h_prev[idx];
    float v  = (1.0f - zg) * ng + zg * h;
    attn_h_s[idx] = v;
    out[off_attnh(*rp) + idx] = v;
}

// ---------------------------------------------------------------------------
// Attention energies: u[b,t] = sum_d tanh(esp[b,t,d] + q[b,d]) * v[d]
// One wave per timestep; 8 channels (two float4) per lane; shuffle reduce.
// ---------------------------------------------------------------------------
__global__ __launch_bounds__(256)
void attn_energy_kernel(const float* __restrict__ esp, const float* __restrict__ q,
                        const float* __restrict__ vvec, float* __restrict__ u)
{
    const int b    = blockIdx.x;
    const int wave = threadIdx.x >> 5;
    const int lane = threadIdx.x & 31;
    const int t    = blockIdx.y * 8 + wave;

    const float4* row = (const float4*)(esp + ((size_t)b * Tt + t) * Dd + lane * 8);
    const float4* qp  = (const float4*)(q    + (size_t)b * Dd + lane * 8);
    const float4* vp  = (const float4*)(vvec + lane * 8);

    float4 e0 = row[0], e1 = row[1];
    float4 q0 = qp[0],  q1 = qp[1];
    float4 v0 = vp[0],  v1 = vp[1];

    float s = 0.0f;
    s += tanh_fast(e0.x + q0.x) * v0.x;
    s += tanh_fast(e0.y + q0.y) * v0.y;
    s += tanh_fast(e0.z + q0.z) * v0.z;
    s += tanh_fast(e0.w + q0.w) * v0.w;
    s += tanh_fast(e1.x + q1.x) * v1.x;
    s += tanh_fast(e1.y + q1.y) * v1.y;
    s += tanh_fast(e1.z + q1.z) * v1.z;
    s += tanh_fast(e1.w + q1.w) * v1.w;

    #pragma unroll
    for (int off = 16; off > 0; off >>= 1)
        s += __shfl_xor(s, off, 32);

    if (lane == 0) u[(size_t)b * Tt + t] = s;
}

// ---------------------------------------------------------------------------
// Softmax over T per batch row; writes scratch copy + output region.
// ---------------------------------------------------------------------------
__global__ __launch_bounds__(256)
void softmax_kernel(const float* __restrict__ u, float* __restrict__ scores_s,
                    float* __restrict__ out, const int* __restrict__ rp)
{
    __shared__ float red[256];
    const int b = blockIdx.x, tid = threadIdx.x;
    const float* ub = u + (size_t)b * Tt;

    float m = -INFINITY;
    for (int t = tid; t < Tt; t += 256) m = fmaxf(m, ub[t]);
    red[tid] = m; __syncthreads();
    for (int s = 128; s > 0; s >>= 1) { if (tid < s) red[tid] = fmaxf(red[tid], red[tid + s]); __syncthreads(); }
    m = red[0]; __syncthreads();

    float sum = 0.0f;
    for (int t = tid; t < Tt; t += 256) sum += expf(ub[t] - m);
    red[tid] = sum; __syncthreads();
    for (int s = 128; s > 0; s >>= 1) { if (tid < s) red[tid] += red[tid + s]; __syncthreads(); }
    float inv = 1.0f / red[0];

    float* outp = out + off_scores(*rp) + (size_t)b * Tt;
    float* ssp  = scores_s + (size_t)b * Tt;
    for (int t = tid; t < Tt; t += 256) {
        float v = expf(ub[t] - m) * inv;
        ssp[t] = v; outp[t] = v;
    }
}

// ---------------------------------------------------------------------------
// Context: ctx[b,d] = sum_t scores[b,t] * enc[b,t,d]  (coalesced stream)
// ---------------------------------------------------------------------------
__global__ __launch_bounds__(256)
void context_kernel(const float* __restrict__ enc, const float* __restrict__ scores_s,
                    float* __restrict__ ctx_s, float* __restrict__ out,
                    const int* __restrict__ rp)
{
    __shared__ float s_lds[Tt];
    const int b = blockIdx.x, d = threadIdx.x;
    for (int t = d; t < Tt; t += 256) s_lds[t] = scores_s[(size_t)b * Tt + t];
    __syncthreads();

    const float* eb = enc + (size_t)b * Tt * Dd + d;
    float acc = 0.0f;
    #pragma unroll 4
    for (int t = 0; t < Tt; ++t) {
        if ((t & 15) == 0) __builtin_prefetch(eb + (size_t)(t + 32) * Dd, 0, 0);
        acc += s_lds[t] * eb[(size_t)t * Dd];
    }
    ctx_s[(size_t)b * Dd + d] = acc;
    out[off_ctx(*rp) + (size_t)b * Dd + d] = acc;
}

// ---------------------------------------------------------------------------
// LSTM gate fusion. which = 0 (rnn1) or 1 (rnn2) selects output regions.
// ---------------------------------------------------------------------------
__global__ __launch_bounds__(256)
void lstm_gate_kernel(const float* __restrict__ g, const float* __restrict__ c_prev,
                      float* __restrict__ h_s, float* __restrict__ c_s,
                      float* __restrict__ out, const int* __restrict__ rp, int which)
{
    int idx = blockIdx.x * blockDim.x + threadIdx.x;   // b*512 + j
    if (idx >= Bz * Ll) return;
    int b = idx >> 9, j = idx & 511;
    const float* gb = g + (size_t)b * (4 * Ll);
    float ig = sigmoidf_(gb[j]);
    float fg = sigmoidf_(gb[Ll + j]);
    float gg = tanhf    (gb[2 * Ll + j]);
    float og = sigmoidf_(gb[3 * Ll + j]);
    float c  = fg * c_prev[idx] + ig * gg;
    float h  = og * tanhf(c);
    h_s[idx] = h; c_s[idx] = c;
    size_t base = off_hc(*rp);
    out[base + (size_t)which * (Bz * Ll) + idx]       = h;   // rnn{1,2}_h
    out[base + (size_t)(2 + which) * (Bz * Ll) + idx] = c;   // rnn{1,2}_c
}

__global__ __launch_bounds__(256)
void add_inplace_kernel(float* __restrict__ x, const float* __restrict__ y, int n)
{
    int idx = blockIdx.x * blockDim.x + threadIdx.x;
    if (idx < n) x[idx] += y[idx];
}

// mels = mel_full.reshape(B,80,20)[:, :, :r]  ->  out[b*80*r + mel*r + j]
__global__ __launch_bounds__(256)
void mel_gather_kernel(const float* __restrict__ mel_full, float* __restrict__ out,
                       const int* __restrict__ rp)
{
    int idx = blockIdx.x * blockDim.x + threadIdx.x;   // b*1600 + mel*20 + j
    if (idx >= Bz * NMELS * MAXR) return;
    int r = *rp;
    int j = idx % MAXR;
    if (j >= r) return;
    int bm = idx / MAXR;                               // b*80 + mel
    out[(size_t)bm * r + j] = mel_full[idx];
}

// ---------------------------------------------------------------------------
extern "C" void kernel_launch(void* const* d_in, const int* in_sizes, int n_in,
                              void* d_out, int out_size, void* d_ws, size_t ws_size,
                              hipStream_t stream)
{
    const float* enc       = (const float*)d_in[0];
    const float* esp       = (const float*)d_in[1];
    const float* prenet_in = (const float*)d_in[2];
    const float* attn_hid  = (const float*)d_in[3];
    const float* rnn1_hid  = (const float*)d_in[4];
    const float* rnn2_hid  = (const float*)d_in[5];
    const float* rnn1_cell = (const float*)d_in[6];
    const float* rnn2_cell = (const float*)d_in[7];
    const float* ctx_vec   = (const float*)d_in[8];
    const float* fc1_W  = (const float*)d_in[9];
    const float* fc1_b  = (const float*)d_in[10];
    const float* fc2_W  = (const float*)d_in[11];
    const float* fc2_b  = (const float*)d_in[12];
    const float* attn_W = (const float*)d_in[13];
    const float* attn_v = (const float*)d_in[14];
    const float* gWih   = (const float*)d_in[15];
    const float* gWhh   = (const float*)d_in[16];
    const float* gbih   = (const float*)d_in[17];
    const float* gbhh   = (const float*)d_in[18];
    const float* riW    = (const float*)d_in[19];
    const float* rib    = (const float*)d_in[20];
    const float* l1Wih  = (const float*)d_in[21];
    const float* l1Whh  = (const float*)d_in[22];
    const float* l1bih  = (const float*)d_in[23];
    const float* l1bhh  = (const float*)d_in[24];
    const float* l2Wih  = (const float*)d_in[25];
    const float* l2Whh  = (const float*)d_in[26];
    const float* l2bih  = (const float*)d_in[27];
    const float* l2bhh  = (const float*)d_in[28];
    const float* mel_W  = (const float*)d_in[29];
    const int*   rp     = (const int*)d_in[30];

    float* out = (float*)d_out;

    // scratch layout (floats)
    float* ws = (float*)d_ws;
    float* p1       = ws;                    // 128*256
    float* p2       = p1 + Bz * 256;         // 128*128
    float* gi       = p2 + Bz * 128;         // 128*768
    float* gh       = gi + Bz * 768;         // 128*768
    float* attn_h_s = gh + Bz * 768;         // 128*256
    float* q        = attn_h_s + Bz * Dd;    // 128*256
    float* u        = q + Bz * Dd;           // 128*1024
    float* scores_s = u + (size_t)Bz * Tt;   // 128*1024
    float* ctx_s    = scores_s + (size_t)Bz * Tt;  // 128*256
    float* x        = ctx_s + Bz * Dd;       // 128*512
    float* h1       = x  + Bz * Ll;          // 128*512
    float* c1       = h1 + Bz * Ll;
    float* h2       = c1 + Bz * Ll;
    float* c2       = h2 + Bz * Ll;
    float* g1       = c2 + Bz * Ll;          // 128*2048
    float* g2       = g1 + (size_t)Bz * 4 * Ll;
    float* mel_full = g2 + (size_t)Bz * 4 * Ll;    // 128*1600

    const dim3 wb(32);
    #define GEMM_GRID(N) dim3((N) / 32, Bz / 16)

    // 1) PreNet fc1: relu(prenet_in @ fc1_W.T + b)      [128,80]->[128,256]
    wmma_gemm2_kernel<<<GEMM_GRID(256), wb, 0, stream>>>(
        prenet_in, NMELS, fc1_W, NMELS, NMELS,
        nullptr, 0, nullptr, 0, 0, fc1_b, nullptr, p1, 256, 1);

    // 2) PreNet fc2: relu(p1 @ fc2_W.T + b)             [128,256]->[128,128]
    wmma_gemm2_kernel<<<GEMM_GRID(128), wb, 0, stream>>>(
        p1, 256, fc2_W, 256, 256,
        nullptr, 0, nullptr, 0, 0, fc2_b, nullptr, p2, 128, 1);

    // 3) GRU gi = [ctx_vec | p2] @ gWih.T + bih         K split 256 + 128
    wmma_gemm2_kernel<<<GEMM_GRID(768), wb, 0, stream>>>(
        ctx_vec, Dd, gWih, 384, Dd,
        p2, 128, gWih + Dd, 384, 128, gbih, nullptr, gi, 768, 0);

    // 4) GRU gh = attn_hidden @ gWhh.T + bhh
    wmma_gemm2_kernel<<<GEMM_GRID(768), wb, 0, stream>>>(
        attn_hid, Dd, gWhh, Dd, Dd,
        nullptr, 0, nullptr, 0, 0, gbhh, nullptr, gh, 768, 0);

    // 5) GRU gates -> attn_h
    gru_gate_kernel<<<(Bz * Dd) / 256, 256, 0, stream>>>(gi, gh, attn_hid, attn_h_s, out, rp);

    // 6) q = attn_h @ attn_W.T
    wmma_gemm2_kernel<<<GEMM_GRID(256), wb, 0, stream>>>(
        attn_h_s, Dd, attn_W, Dd, Dd,
        nullptr, 0, nullptr, 0, 0, nullptr, nullptr, q, Dd, 0);

    // 7) energies u[b,t]   (streams 128MB of encoder_seq_proj)
    attn_energy_kernel<<<dim3(Bz, Tt / 8), 256, 0, stream>>>(esp, q, attn_v, u);

    // 8) softmax over t -> scores (scratch + output)
    softmax_kernel<<<Bz, 256, 0, stream>>>(u, scores_s, out, rp);

    // 9) context = scores @ encoder_seq   (streams 128MB of encoder_seq)
    context_kernel<<<Bz, 256, 0, stream>>>(enc, scores_s, ctx_s, out, rp);

    // 10) x = [context | attn_h] @ rnn_in_W.T + b       K split 256 + 256
    wmma_gemm2_kernel<<<GEMM_GRID(Ll), wb, 0, stream>>>(
        ctx_s, Dd, riW, 2 * Dd, Dd,
        attn_h_s, Dd, riW + Dd, 2 * Dd, Dd, rib, nullptr, x, Ll, 0);

    // 11) LSTM1 gates: g1 = x @ Wih.T + h @ Whh.T + bih + bhh
    wmma_gemm2_kernel<<<GEMM_GRID(4 * Ll), wb, 0, stream>>>(
        x, Ll, l1Wih, Ll, Ll,
        rnn1_hid, Ll, l1Whh, Ll, Ll, l1bih, l1bhh, g1, 4 * Ll, 0);
    lstm_gate_kernel<<<(Bz * Ll) / 256, 256, 0, stream>>>(g1, rnn1_cell, h1, c1, out, rp, 0);
    add_inplace_kernel<<<(Bz * Ll) / 256, 256, 0, stream>>>(x, h1, Bz * Ll);

    // 12) LSTM2
    wmma_gemm2_kernel<<<GEMM_GRID(4 * Ll), wb, 0, stream>>>(
        x, Ll, l2Wih, Ll, Ll,
        rnn2_hid, Ll, l2Whh, Ll, Ll, l2bih, l2bhh, g2, 4 * Ll, 0);
    lstm_gate_kernel<<<(Bz * Ll) / 256, 256, 0, stream>>>(g2, rnn2_cell, h2, c2, out, rp, 1);
    add_inplace_kernel<<<(Bz * Ll) / 256, 256, 0, stream>>>(x, h2, Bz * Ll);

    // 13) mel projection + dynamic-r gather
    wmma_gemm2_kernel<<<GEMM_GRID(NMELS * MAXR), wb, 0, stream>>>(
        x, Ll, mel_W, Ll, Ll,
        nullptr, 0, nullptr, 0, 0, nullptr, nullptr, mel_full, NMELS * MAXR, 0);
    mel_gather_kernel<<<(Bz * NMELS * MAXR + 255) / 256, 256, 0, stream>>>(mel_full, out, rp);
}